// VoronoiDecoder_64355789964118
// MI455X (gfx1250) — compile-verified
//
#include <hip/hip_runtime.h>
#include <hip/hip_bf16.h>
#include <math.h>
#include <stdint.h>

// ---------------------------------------------------------------------------
// VoronoiDecoder fused kernel for gfx1250 (MI455X, wave32, WMMA)
//
// d^2[q,s] = sum_k Fq[q,k]*Fs[s,k]  (rank-6 bilinear form, K padded to 8)
//   -> V_WMMA_F32_16X16X4_F32 x2 per 16(seed) x 16(query) tile
// C layout: lane = N(query)%16 + 16*half, VGPR r = M(seed) = r + 8*half
//   -> per-lane flash-softmax + top-2 over the seed axis, one shfl_xor(16)
//      merge at the end. Distance matrix never touches memory.
// Round-1 fix: raw v_sqrt_f32 / v_rcp_f32 instead of IEEE fixup sequences
// (operands are O(1) distances, nowhere near denormals) -> ~25% fewer VALU
// ops in the hot loop that competes with the exp/select work.
// ---------------------------------------------------------------------------

typedef float v2f __attribute__((ext_vector_type(2)));
typedef float v8f __attribute__((ext_vector_type(8)));

#define EPS        1e-8f
#define W_MIN      0.005f
#define INV_BETA   50.0f      // 1/0.02
#define BETA       0.02f
#define INV_RAWT   0.2f       // 1/5.0
#define J_K0       3.0f
#define USE_ASYNC_LDS 1

#define QQ 65536
#define SS 512
#define QPB 128               // queries per block (8 waves * 16)

__device__ __forceinline__ float fsqrt_raw(float x) {
#if defined(__gfx1250__)
  return __builtin_amdgcn_sqrtf(x);      // single v_sqrt_f32
#else
  return sqrtf(x);
#endif
}

__device__ __forceinline__ float frcp_raw(float x) {
#if defined(__gfx1250__)
  return __builtin_amdgcn_rcpf(x);       // single v_rcp_f32
#else
  return 1.0f / x;
#endif
}

__device__ __forceinline__ float sigf(float x) {
  return frcp_raw(1.0f + __expf(-x));
}

__device__ __forceinline__ v8f wmma_f32_16x16x4(v2f a, v2f b, v8f c) {
#if defined(__gfx1250__)
  // 8 args: (neg_a, A, neg_b, B, c_mod, C, reuse_a, reuse_b)
  return __builtin_amdgcn_wmma_f32_16x16x4_f32(false, a, false, b, (short)0, c,
                                               false, false);
#else
  (void)a; (void)b;
  return c;   // host-pass stub, never executed
#endif
}

// ---- kernel 1: seed features  Fs[s] = [A,B,C,P,Q,R,0,0] -> d_ws (512*8 f32)
__global__ __launch_bounds__(256) void voronoi_seedfeat(
    const float* __restrict__ seeds, const float* __restrict__ theta,
    const float* __restrict__ a_raw, float* __restrict__ feat) {
  int s = blockIdx.x * 256 + threadIdx.x;
  if (s >= SS) return;
  float px = seeds[2 * s + 0];
  float py = seeds[2 * s + 1];
  float th = theta[s];
  float a  = 0.5f + 1.5f * sigf(a_raw[s]);
  float c  = __cosf(th), sn = __sinf(th);
  float a2 = a * a, ia2 = frcp_raw(a2);
  float A = a2 * c * c + sn * sn * ia2;
  float B = 2.0f * c * sn * (a2 - ia2);
  float C = a2 * sn * sn + c * c * ia2;
  float P = -(2.0f * A * px + B * py);
  float Q = -(B * px + 2.0f * C * py);
  float R = A * px * px + B * px * py + C * py * py;
  float* f = feat + 8 * s;
  f[0] = A; f[1] = B; f[2] = C; f[3] = P;
  f[4] = Q; f[5] = R; f[6] = 0.0f; f[7] = 0.0f;
}

// ---- kernel 2: fused distance-GEMM + flash softmax + top2 + epilogue
__global__ __launch_bounds__(256) void voronoi_main(
    const float* __restrict__ uv, const float* __restrict__ seeds,
    const float* __restrict__ w_raw, const float* __restrict__ feat,
    const float* __restrict__ h_raw, const float* __restrict__ gap_thr_raw,
    const float* __restrict__ big_thr_raw, const float* __restrict__ alpha_raw,
    const float* __restrict__ eta_raw, float* __restrict__ out) {
  __shared__ __align__(16) float sFeat[SS * 8];   // 16 KB

  const int tid   = threadIdx.x;
  const int wave  = tid >> 5;
  const int lane  = tid & 31;
  const int half  = lane >> 4;     // 0: K 0,1,4,5 / M 0..7 ; 1: K 2,3,6,7 / M 8..15
  const int qlane = lane & 15;

  // ---- stage seed-feature table into LDS (CDNA5 async global->LDS) --------
#if defined(__gfx1250__) && USE_ASYNC_LDS
  {
    unsigned lbase = (unsigned)(uint64_t)(uintptr_t)&sFeat[0];
    #pragma unroll
    for (int i = 0; i < 4; ++i) {
      unsigned goff = (unsigned)(tid * 16 + i * 4096);
      unsigned loff = lbase + (unsigned)(tid * 16 + i * 4096);
      asm volatile("global_load_async_to_lds_b128 %0, %1, %2"
                   :
                   : "v"(loff), "v"(goff), "s"(feat)
                   : "memory");
    }
    asm volatile("s_wait_asynccnt 0" ::: "memory");
  }
#else
  for (int i = tid; i < SS * 8; i += 256) sFeat[i] = feat[i];
#endif
  __syncthreads();

  // ---- query features, pre-selected into WMMA B-operand slices ------------
  const int q = blockIdx.x * QPB + wave * 16 + qlane;
  const float u = uv[2 * q + 0];
  const float v = uv[2 * q + 1];
  const float f0 = u * u, f1 = u * v, f2 = v * v;
  const float f3 = u, f4 = v, f5 = 1.0f;
  // B (KxN): VGPR0 = rows K0|K2, VGPR1 = rows K1|K3, N = lane%16
  v2f b1, b2;
  b1.x = half ? f2 : f0;  b1.y = half ? f3 : f1;
  b2.x = half ? 0.f : f4; b2.y = half ? 0.f : f5;

  const v2f* ldsF2 = (const v2f*)sFeat;

  // ---- online state: top-2 + softmax sums (centered at running min d1) ----
  float d1 = 3.0e38f, d2 = 3.0e38f;
  int   i1 = 0, i2 = 0;
  float s1 = 0.0f, s2 = 0.0f;

  #pragma unroll 2
  for (int t = 0; t < SS / 16; ++t) {
    // A (MxK): lane m<16 -> K0,K1 ; lane m+16 -> K2,K3 (then K4..7)
    const int fb2 = (t * 16 + qlane) * 4;
    v2f a1 = ldsF2[fb2 + half];
    v2f a2 = ldsF2[fb2 + 2 + half];
    v8f c = {};
    c = wmma_f32_16x16x4(a1, b1, c);
    c = wmma_f32_16x16x4(a2, b2, c);

    const int sbase = t * 16 + half * 8;
    #pragma unroll
    for (int r = 0; r < 8; ++r) {
      float d   = fsqrt_raw(c[r] + EPS);
      int   sid = sbase + r;
      // exp(-|d - d1|/beta): works as both tail term and rescale factor
      float e  = __expf((fminf(d, d1) - fmaxf(d, d1)) * INV_BETA);
      float e2 = e * e;
      bool  nm = d < d1;
      s1 = nm ? __fmaf_rn(s1, e, 1.0f)  : (s1 + e);
      s2 = nm ? __fmaf_rn(s2, e2, 1.0f) : (s2 + e2);
      i2 = nm ? i1 : ((d < d2) ? sid : i2);
      d2 = nm ? d1 : fminf(d2, d);
      i1 = nm ? sid : i1;
      d1 = nm ? d   : d1;
    }
  }

  // ---- merge lane L with lane L^16 (same query, other 8 seed rows) --------
  float d1b = __shfl_xor(d1, 16), d2b = __shfl_xor(d2, 16);
  float s1b = __shfl_xor(s1, 16), s2b = __shfl_xor(s2, 16);
  int   i1b = __shfl_xor(i1, 16), i2b = __shfl_xor(i2, 16);

  float mmin = fminf(d1, d1b);
  float fa = __expf((mmin - d1)  * INV_BETA);
  float fb = __expf((mmin - d1b) * INV_BETA);
  float S1 = s1 * fa + s1b * fb;
  float S2 = s2 * fa * fa + s2b * fb * fb;

  float D1, rest; int I1, restI;
  if (d1 <= d1b) { D1 = d1;  I1 = i1;  rest = d1b; restI = i1b; }
  else           { D1 = d1b; I1 = i1b; rest = d1;  restI = i1;  }
  float D2; int I2;
  if (d2 <= d2b) { D2 = d2;  I2 = i2;  } else { D2 = d2b; I2 = i2b; }
  if (rest < D2) { D2 = rest; I2 = restI; }

  if ((lane & 16) == 0) {
    // ---- epilogue (per query) ---------------------------------------------
    float gap  = D2 - D1;
    float keff = S1 * S1 * frcp_raw(S2 + EPS * S1 * S1);  // 1/(sum p^2 + eps)
    float jfac = sigf((keff - J_K0) * (1.0f / 0.35f));

    // w_pair = 0.5*(w_geo[i,j] + w_geo[j,i])
    float pix = seeds[2 * I1 + 0], piy = seeds[2 * I1 + 1];
    float pjx = seeds[2 * I2 + 0], pjy = seeds[2 * I2 + 1];
    float ddx = pix - pjx, ddy = piy - pjy;
    float dist = fsqrt_raw(ddx * ddx + ddy * ddy + EPS);
    float wmax = fmaxf(0.8f * dist, W_MIN + EPS);
    float wij  = w_raw[I1 * SS + I2];
    float wji  = w_raw[I2 * SS + I1];
    float g1 = W_MIN + (wmax - W_MIN) * sigf(wij * INV_RAWT);
    float g2 = W_MIN + (wmax - W_MIN) * sigf(wji * INV_RAWT);
    float w_pair = 0.5f * (g1 + g2);

    float w_eff    = w_pair * (1.0f + 0.15f * jfac);
    float beta_eff = BETA * (1.0f + 0.05f * jfac);
    float wall     = sigf((w_eff - gap) * frcp_raw(beta_eff));

    float h       = 0.5f + 1.5f * sigf(h_raw[0]);
    float gap_thr = 0.5f * sigf(gap_thr_raw[0]);
    float big_thr = 0.6f * sigf(big_thr_raw[0]);
    float alpha_g = 0.01f + 0.19f * sigf(alpha_raw[0]);
    float eta_g   = 0.01f + 0.19f * sigf(eta_raw[0]);

    float gate = sigf((gap_thr - gap) * frcp_raw(alpha_g)) *
                 sigf((big_thr - D1) * frcp_raw(eta_g));
    float m3   = fmaxf(keff - J_K0, 0.0f);
    float trip = 0.15f * m3 * fsqrt_raw(m3);      // m3^1.5
    float occ  = fminf(fmaxf(wall * gate + trip, 0.0f), 1.0f);

    out[q] = occ * h;
  }
}

extern "C" void kernel_launch(void* const* d_in, const int* in_sizes, int n_in,
                              void* d_out, int out_size, void* d_ws, size_t ws_size,
                              hipStream_t stream) {
  const float* uv          = (const float*)d_in[0];
  const float* seeds_raw   = (const float*)d_in[1];
  const float* w_raw       = (const float*)d_in[2];
  const float* h_raw       = (const float*)d_in[3];
  const float* theta       = (const float*)d_in[4];
  const float* a_raw       = (const float*)d_in[5];
  const float* gap_thr_raw = (const float*)d_in[6];
  const float* big_thr_raw = (const float*)d_in[7];
  const float* alpha_raw   = (const float*)d_in[8];
  const float* eta_raw     = (const float*)d_in[9];
  float* out  = (float*)d_out;
  float* feat = (float*)d_ws;                 // 512*8 f32 = 16 KB scratch

  voronoi_seedfeat<<<2, 256, 0, stream>>>(seeds_raw, theta, a_raw, feat);
  voronoi_main<<<QQ / QPB, 256, 0, stream>>>(uv, seeds_raw, w_raw, feat,
                                             h_raw, gap_thr_raw, big_thr_raw,
                                             alpha_raw, eta_raw, out);
}